// AdaptiveRankingLoss_71889162601016
// MI455X (gfx1250) — compile-verified
//
#include <hip/hip_runtime.h>

#define TILE 256
#define JSLICES 32

typedef float v2f __attribute__((ext_vector_type(2)));
typedef float v8f __attribute__((ext_vector_type(8)));

// Main pair-loop kernel. Computes over ALL ordered pairs (i, j): the pair
// contribution is symmetric under i<->j, and i==j self-masks (tdiff==0), so
// loss_all = 2*loss_triu and count_all = 2*count_triu; the final ratio is
// identical to the reference's triu version.
__global__ __launch_bounds__(TILE) void AdaptiveRankingLoss_pairs_kernel(
    const float* __restrict__ pred,
    const float* __restrict__ targ,
    const float* __restrict__ unc,
    float2* __restrict__ partials,   // one (loss, count) per block
    int n, int numIBlocks, int jslices)
{
    __shared__ float sp[TILE];
    __shared__ float st[TILE];
    __shared__ float su[TILE];
    __shared__ float waveRed[TILE / 32][4];

    const int tid = threadIdx.x;
    const int bi  = blockIdx.x % numIBlocks;   // which i-tile
    const int js  = blockIdx.x / numIBlocks;   // which j-slice (strided tiles)
    const int i   = bi * TILE + tid;

    const bool ivalid = (i < n);
    float pi = 0.0f, ti = 0.0f, ui1 = 1.0f;
    if (ivalid) {
        pi  = pred[i];
        ti  = targ[i];
        ui1 = 1.0f + unc[i];
    }

    float loss = 0.0f;
    float cnt  = 0.0f;

    const int numTiles = (n + TILE - 1) / TILE;
    for (int t = js; t < numTiles; t += jslices) {
        const int base = t * TILE;
        const int g    = base + tid;
        const bool v   = (g < n);
        sp[tid] = v ? pred[g] : 0.0f;
        st[tid] = v ? targ[g] : 0.0f;
        su[tid] = v ? unc[g]  : 0.0f;
        __syncthreads();

        const int m = (n - base) < TILE ? (n - base) : TILE;
        if (ivalid) {
            #pragma unroll 8
            for (int k = 0; k < m; ++k) {
                const float td     = ti - st[k];
                const float pd     = pi - sp[k];
                const float atd    = __builtin_fabsf(td);
                const float margin = 0.1f * fminf(fmaxf(atd, 0.1f), 1.0f);
                // sign(td)*(-pd): td==0 case is masked below, so copysign is fine
                const float s      = __builtin_copysignf(1.0f, td);
                const float h      = fmaxf(__builtin_fmaf(-s, pd, margin), 0.0f);
                // denominator in [1,3]; v_rcp_f32 is ~1 ulp accurate -> skip
                // the IEEE-div Newton refinement chain (biggest VALU cost here)
                const float w      = __builtin_amdgcn_rcpf(ui1 + su[k]);
                const bool  mask   = (td != 0.0f);
                loss += mask ? (w * h) : 0.0f;
                cnt  += mask ? 1.0f : 0.0f;
            }
        }
        __syncthreads();
    }

    // ---- wave32 reduction via V_WMMA_F32_16X16X4_F32 ----
    // A (16x4, 2 VGPRs/lane): every lane contributes {loss, cnt}.
    //   A[m][0]=loss(lane m), A[m][1]=cnt(lane m),
    //   A[m][2]=loss(lane m+16), A[m][3]=cnt(lane m+16)
    // B (4x16) selector: column 0 <- rows {0,2} (loss), column 1 <- rows {1,3} (cnt).
    //   B VGPR0: lanes 0-15 = row K=0, lanes 16-31 = row K=1
    //   B VGPR1: lanes 0-15 = row K=2, lanes 16-31 = row K=3
    // => lanes<16: b = (col==0); lanes>=16: b = (col==1)
    const int  lane = tid & 31;
    const int  col  = lane & 15;
    const bool hi   = (lane >= 16);
    const float bsel = (col == (hi ? 1 : 0)) ? 1.0f : 0.0f;

    v2f a; a.x = loss; a.y = cnt;
    v2f b; b.x = bsel; b.y = bsel;
    v8f c = {};
    v8f d = __builtin_amdgcn_wmma_f32_16x16x4_f32(
        /*neg_a=*/false, a, /*neg_b=*/false, b,
        /*c_mod=*/(short)0, c, /*reuse_a=*/false, /*reuse_b=*/false);

    // lane 0:  D[0..7][0]  = loss partial (rows 0-7)
    // lane 16: D[8..15][0] = loss partial (rows 8-15)
    // lane 1:  D[0..7][1]  = cnt partial
    // lane 17: D[8..15][1] = cnt partial
    const float s8 = d[0] + d[1] + d[2] + d[3] + d[4] + d[5] + d[6] + d[7];

    const int w = tid >> 5;
    if (lane == 0)  waveRed[w][0] = s8;
    if (lane == 16) waveRed[w][1] = s8;
    if (lane == 1)  waveRed[w][2] = s8;
    if (lane == 17) waveRed[w][3] = s8;
    __syncthreads();

    if (tid == 0) {
        float L = 0.0f, C = 0.0f;
        #pragma unroll
        for (int ww = 0; ww < TILE / 32; ++ww) {
            L += waveRed[ww][0] + waveRed[ww][1];
            C += waveRed[ww][2] + waveRed[ww][3];
        }
        partials[blockIdx.x] = make_float2(L, C);
    }
}

// Deterministic final reduction in double precision, then loss / max(count, 1).
__global__ __launch_bounds__(256) void AdaptiveRankingLoss_finalize_kernel(
    const float2* __restrict__ partials, int numPartials, float* __restrict__ out)
{
    __shared__ double sL[256];
    __shared__ double sC[256];
    double L = 0.0, C = 0.0;
    for (int idx = threadIdx.x; idx < numPartials; idx += 256) {
        const float2 pc = partials[idx];
        L += (double)pc.x;
        C += (double)pc.y;
    }
    sL[threadIdx.x] = L;
    sC[threadIdx.x] = C;
    __syncthreads();
    for (int s = 128; s > 0; s >>= 1) {
        if (threadIdx.x < s) {
            sL[threadIdx.x] += sL[threadIdx.x + s];
            sC[threadIdx.x] += sC[threadIdx.x + s];
        }
        __syncthreads();
    }
    if (threadIdx.x == 0) {
        const double cden = (sC[0] < 1.0) ? 1.0 : sC[0];
        out[0] = (float)(sL[0] / cden);
    }
}

extern "C" void kernel_launch(void* const* d_in, const int* in_sizes, int n_in,
                              void* d_out, int out_size, void* d_ws, size_t ws_size,
                              hipStream_t stream)
{
    const float* pred = (const float*)d_in[0];
    const float* targ = (const float*)d_in[1];
    const float* unc  = (const float*)d_in[2];
    const int n = in_sizes[0];

    const int numIBlocks = (n + TILE - 1) / TILE;

    // Clamp j-slices so the (loss,cnt) partials fit in workspace.
    int jslices = JSLICES;
    while (jslices > 1 &&
           (size_t)numIBlocks * (size_t)jslices * sizeof(float2) > ws_size) {
        jslices >>= 1;
    }
    const int numBlocks = numIBlocks * jslices;

    float2* partials = (float2*)d_ws;

    AdaptiveRankingLoss_pairs_kernel<<<numBlocks, TILE, 0, stream>>>(
        pred, targ, unc, partials, n, numIBlocks, jslices);
    AdaptiveRankingLoss_finalize_kernel<<<1, 256, 0, stream>>>(
        partials, numBlocks, (float*)d_out);
}